// RuleAwareProjection_7232724926953
// MI455X (gfx1250) — compile-verified
//
#include <hip/hip_runtime.h>
#include <stdint.h>

typedef float v2f __attribute__((ext_vector_type(2)));
typedef float v4f __attribute__((ext_vector_type(4)));
typedef float v8f __attribute__((ext_vector_type(8)));
typedef unsigned int v4u __attribute__((ext_vector_type(4)));
typedef int v4i __attribute__((ext_vector_type(4)));
typedef int v8i __attribute__((ext_vector_type(8)));

#define NTOK   16384
#define INF    2048
#define OUTF   2048
#define RANK   45
#define RANKP  48
#define ADIM   32   // a (X cols, U cols)
#define BDIM   64   // b (X rows, V cols)
#define CDIM   32   // c (U rows, out minor)
#define DDIM   64   // d (V rows, out major)

#define TOKS   16          // tokens per block
#define WAVES  8           // 256 threads
#define XS_STRIDE 2052     // 2048 + 4 pad (bank-conflict-free token-major reads)
#define TS_STRIDE 49       // 48 + 1 pad
#define C1_STRIDE 52       // 8*52 % 64 = 32 (store), 2*52 % 64 = 40 (load): conflict-free

#define XS_FLOATS (TOKS * XS_STRIDE)
#define TS_FLOATS (TOKS * TS_STRIDE)
#define C1_FLOATS (WAVES * BDIM * C1_STRIDE)
#define LDS_BYTES ((XS_FLOATS + TS_FLOATS + C1_FLOATS) * 4)

#if defined(__has_builtin)
#if __has_builtin(__builtin_amdgcn_tensor_load_to_lds) && \
    __has_builtin(__builtin_amdgcn_s_wait_tensorcnt)
#define HAVE_TDM 1
#endif
#endif
#ifndef HAVE_TDM
#define HAVE_TDM 0
#endif

static __device__ __forceinline__ v8f wmma_f32(v2f a, v2f b, v8f c) {
    // D = A(16x4, f32) * B(4x16, f32) + C(16x16, f32)
    return __builtin_amdgcn_wmma_f32_16x16x4_f32(false, a, false, b, (short)0, c,
                                                 false, false);
}

__global__ __launch_bounds__(256, 1)
void rule_proj_kernel(const float* __restrict__ x,      // [N, 2048]
                      const int*   __restrict__ rid,    // [N]
                      const float* __restrict__ Win,    // [2048, 45]
                      const float* __restrict__ Wout,   // [45, 2048]
                      const float* __restrict__ Ur,     // [64, 32, 32] (r, c, a)
                      const float* __restrict__ Vr,     // [64, 64, 64] (r, d, b)
                      const float* __restrict__ gain,   // [64]
                      float*       __restrict__ out)    // [N, 2048]
{
    extern __shared__ float smem[];
    float* xs = smem;                    // phase 0/1: staged x rows; phase 2+: base out (bs)
    float* ts = smem + XS_FLOATS;        // [16][49] rank intermediate (padded 45->48)
    float* c1 = ts + TS_FLOATS;          // per-wave [64][52] C1 scratch

    const int tid   = threadIdx.x;
    const int wave  = tid >> 5;
    const int lane  = tid & 31;
    const int lhalf = lane >> 4;         // 0 or 1
    const int lmod  = lane & 15;
    const int tok0  = blockIdx.x * TOKS;

    const v8f zero8 = {};

    // ---------------- Phase 0: stage x tile into LDS, zero ts ----------------
#if HAVE_TDM
    // Tensor Data Mover: one 1x2048 fp32 tile per token row (2 rows per wave).
    // D# group0: count=1 | lds_addr[63:32] | global_addr[120:64] | type=2[127:126]
    // D# group1: data_size=4B, tensor_dim0=2048, tensor_dim1=1, tile_dim0=2048,
    //            tile_dim1=1, tensor_dim0_stride=2048, no pad / no multicast.
    for (int i = 0; i < 2; ++i) {
        const int row = wave * 2 + i;
        const uint64_t ga = (uint64_t)(uintptr_t)(x + (size_t)(tok0 + row) * INF);
        const uint32_t la = (uint32_t)(uintptr_t)(xs + row * XS_STRIDE);
        const uint32_t ga_lo = (uint32_t)__builtin_amdgcn_readfirstlane((int)(uint32_t)ga);
        const uint32_t ga_hi = (uint32_t)__builtin_amdgcn_readfirstlane((int)(uint32_t)(ga >> 32));
        const uint32_t la_u  = (uint32_t)__builtin_amdgcn_readfirstlane((int)la);

        v4u g0;
        g0.x = 1u;                                        // count=1, user mode
        g0.y = la_u;                                      // lds_addr
        g0.z = ga_lo;                                     // global_addr[31:0]
        g0.w = (ga_hi & 0x01FFFFFFu) | (2u << 30);        // global_addr[56:32] | type=2

        v8i g1;
        g1[0] = 0x00020000;                               // data_size=2 (4 bytes)
        g1[1] = (int)(2048u << 16);                       // tensor_dim0[15:0] @ [63:48]
        g1[2] = (int)(1u << 16);                          // tensor_dim0 hi=0, tensor_dim1=1
        g1[3] = (int)(2048u << 16);                       // tile_dim0=2048 @ [127:112]
        g1[4] = 1;                                        // tile_dim1=1, tile_dim2=0
        g1[5] = 2048;                                     // tensor_dim0_stride[31:0]
        g1[6] = 0;                                        // strides hi
        g1[7] = 0;

        v4i gz = {};
#if __clang_major__ >= 23
        v8i z8 = {};
        __builtin_amdgcn_tensor_load_to_lds(g0, g1, gz, gz, z8, 0);
#else
        __builtin_amdgcn_tensor_load_to_lds(g0, g1, gz, gz, 0);
#endif
    }
    __builtin_amdgcn_s_wait_tensorcnt(0);
#else
    for (int i = tid; i < TOKS * (INF / 4); i += 256) {
        int row  = i >> 9;               // / 512
        int col4 = i & 511;
        v4f v = *(const v4f*)(x + (size_t)(tok0 + row) * INF + col4 * 4);
        *(v4f*)(xs + row * XS_STRIDE + col4 * 4) = v;
    }
#endif
    for (int i = tid; i < TS_FLOATS; i += 256) ts[i] = 0.0f;
    __syncthreads();

    // ------------- Phase 1: t[16x48] = xs[16x2048] @ Win[2048x45] -------------
    // K split across 8 waves (256 each); partials merged with ds_add_f32.
    // Rank padding handled branchlessly: clamped address + multiply-by-mask.
    {
        int   nclamp[3];
        float nmask[3];
#pragma unroll
        for (int nt = 0; nt < 3; ++nt) {
            const int n = nt * 16 + lmod;
            nclamp[nt] = (n < RANK) ? n : (RANK - 1);
            nmask[nt]  = (n < RANK) ? 1.0f : 0.0f;
        }
        v8f acc[3];
        acc[0] = zero8; acc[1] = zero8; acc[2] = zero8;
        const int kbase = wave * (INF / WAVES);
        for (int ks = 0; ks < (INF / WAVES) / 4; ++ks) {
            const int k = kbase + ks * 4 + lhalf * 2;
            v2f af;
            af.x = xs[lmod * XS_STRIDE + k];
            af.y = xs[lmod * XS_STRIDE + k + 1];
#pragma unroll
            for (int nt = 0; nt < 3; ++nt) {
                v2f bf;
                bf.x = Win[(size_t)k       * RANK + nclamp[nt]] * nmask[nt];
                bf.y = Win[(size_t)(k + 1) * RANK + nclamp[nt]] * nmask[nt];
                acc[nt] = wmma_f32(af, bf, acc[nt]);
            }
        }
#pragma unroll
        for (int nt = 0; nt < 3; ++nt) {
            const int n = nt * 16 + lmod;
#pragma unroll
            for (int v = 0; v < 8; ++v) {
                const int m = v + 8 * lhalf;
                atomicAdd(&ts[m * TS_STRIDE + n], acc[nt][v]);
            }
        }
    }
    __syncthreads();

    // ------------- Phase 2: bs[16x2048] = t[16x48] @ Wout[45x2048] -------------
    // 128 N-tiles split across 8 waves; result overwrites xs space (xs is dead).
    {
        const int tiles = (OUTF / 16) / WAVES;   // 16
        for (int t = 0; t < tiles; ++t) {
            const int nt = wave * tiles + t;
            const int n  = nt * 16 + lmod;
            v8f acc = zero8;
#pragma unroll
            for (int ks = 0; ks < RANKP / 4; ++ks) {
                const int k   = ks * 4 + lhalf * 2;
                const int kc0 = (k     < RANK) ? k       : (RANK - 1);
                const int kc1 = (k + 1 < RANK) ? (k + 1) : (RANK - 1);
                const float m0 = (k     < RANK) ? 1.0f : 0.0f;
                const float m1 = (k + 1 < RANK) ? 1.0f : 0.0f;
                v2f af;
                af.x = ts[lmod * TS_STRIDE + k];
                af.y = ts[lmod * TS_STRIDE + k + 1];
                v2f bf;
                bf.x = Wout[(size_t)kc0 * OUTF + n] * m0;
                bf.y = Wout[(size_t)kc1 * OUTF + n] * m1;
                acc = wmma_f32(af, bf, acc);
            }
#pragma unroll
            for (int v = 0; v < 8; ++v) {
                const int m = v + 8 * lhalf;
                xs[m * XS_STRIDE + n] = acc[v];      // bs
            }
        }
    }
    __syncthreads();

    // ---- Phase 3: per-token Kronecker adapter, 2 tokens per wave ----
    float* myc1 = c1 + wave * (BDIM * C1_STRIDE);
    for (int tt = 0; tt < 2; ++tt) {
        const int tk = wave * 2 + tt;
        const int g  = tok0 + tk;
        const int r  = rid[g];
        const float gn = gain[r];
        const float* U    = Ur + (size_t)r * CDIM * ADIM;   // U[c][a]
        const float* V    = Vr + (size_t)r * DDIM * BDIM;   // V[d][b]
        const float* xrow = x + (size_t)g * INF;            // X[b][a] = xrow[b*32+a]

        // GEMM1: C1[64x32] = X(64x32) @ U^T(32x32)   (M=b, N=c, K=a)
        v8f acc1[4][2];
#pragma unroll
        for (int mt = 0; mt < 4; ++mt) { acc1[mt][0] = zero8; acc1[mt][1] = zero8; }
#pragma unroll
        for (int ks = 0; ks < ADIM / 4; ++ks) {
            const int k = ks * 4 + lhalf * 2;               // a index
            v2f bf[2];
#pragma unroll
            for (int nt = 0; nt < 2; ++nt) {
                const int c = nt * 16 + lmod;
                bf[nt].x = U[c * ADIM + k];                 // B[a][c] = U[c][a]
                bf[nt].y = U[c * ADIM + k + 1];
            }
#pragma unroll
            for (int mt = 0; mt < 4; ++mt) {
                const int b = mt * 16 + lmod;
                v2f af;
                af.x = xrow[b * ADIM + k];
                af.y = xrow[b * ADIM + k + 1];
                acc1[mt][0] = wmma_f32(af, bf[0], acc1[mt][0]);
                acc1[mt][1] = wmma_f32(af, bf[1], acc1[mt][1]);
            }
        }

        // Spill C1 (C/D layout -> row-major LDS, stride 52, conflict-free both ways)
#pragma unroll
        for (int mt = 0; mt < 4; ++mt)
#pragma unroll
            for (int nt = 0; nt < 2; ++nt)
#pragma unroll
                for (int v = 0; v < 8; ++v) {
                    const int b = mt * 16 + v + 8 * lhalf;
                    const int c = nt * 16 + lmod;
                    myc1[b * C1_STRIDE + c] = acc1[mt][nt][v];
                }

        // GEMM2: C2[64x32] = V(64x64) @ C1(64x32)   (M=d, N=c, K=b)
        v8f acc2[4][2];
#pragma unroll
        for (int mt = 0; mt < 4; ++mt) { acc2[mt][0] = zero8; acc2[mt][1] = zero8; }
#pragma unroll
        for (int ks = 0; ks < BDIM / 4; ++ks) {
            const int k = ks * 4 + lhalf * 2;               // b index
            v2f bf[2];
#pragma unroll
            for (int nt = 0; nt < 2; ++nt) {
                const int c = nt * 16 + lmod;
                bf[nt].x = myc1[k * C1_STRIDE + c];
                bf[nt].y = myc1[(k + 1) * C1_STRIDE + c];
            }
#pragma unroll
            for (int mt = 0; mt < 4; ++mt) {
                const int d = mt * 16 + lmod;
                v2f af;
                af.x = V[d * BDIM + k];
                af.y = V[d * BDIM + k + 1];
                acc2[mt][0] = wmma_f32(af, bf[0], acc2[mt][0]);
                acc2[mt][1] = wmma_f32(af, bf[1], acc2[mt][1]);
            }
        }

        // out[g][d*32+c] = gain * C2[d][c] + bs[tk][d*32+c]
#pragma unroll
        for (int mt = 0; mt < 4; ++mt)
#pragma unroll
            for (int nt = 0; nt < 2; ++nt)
#pragma unroll
                for (int v = 0; v < 8; ++v) {
                    const int d   = mt * 16 + v + 8 * lhalf;
                    const int c   = nt * 16 + lmod;
                    const int col = d * CDIM + c;
                    out[(size_t)g * OUTF + col] =
                        acc2[mt][nt][v] * gn + xs[tk * XS_STRIDE + col];
                }
    }
}

extern "C" void kernel_launch(void* const* d_in, const int* in_sizes, int n_in,
                              void* d_out, int out_size, void* d_ws, size_t ws_size,
                              hipStream_t stream) {
    const float* x    = (const float*)d_in[0];
    const int*   rid  = (const int*)  d_in[1];
    const float* Win  = (const float*)d_in[2];
    const float* Wout = (const float*)d_in[3];
    const float* Ur   = (const float*)d_in[4];
    const float* Vr   = (const float*)d_in[5];
    const float* gn   = (const float*)d_in[6];
    float*       out  = (float*)d_out;

    dim3 grid(NTOK / TOKS);   // 1024 blocks of 16 tokens
    dim3 block(256);          // 8 waves (wave32)
    rule_proj_kernel<<<grid, block, LDS_BYTES, stream>>>(x, rid, Win, Wout, Ur, Vr, gn, out);
}